// PLRNNsyn_88459146428769
// MI455X (gfx1250) — compile-verified
//
#include <hip/hip_runtime.h>
#include <math.h>

#define DZ   128
#define DXR  256
#define TLEN 100000

typedef __attribute__((ext_vector_type(2))) float v2f;
typedef __attribute__((ext_vector_type(8))) float v8f;

// ---------------------------------------------------------------------------
// Phase 1: sequential PLRNN scan. One workgroup, 128 threads (4 wave32s).
// Thread d owns state row d. W1,W2 live in LDS transposed (stride 129 =>
// bank-conflict-free column reads and staging writes). Activation vector is
// broadcast through a 128-float LDS buffer. Writes z_t to ws as [T, DZ].
// ---------------------------------------------------------------------------
__global__ void __launch_bounds__(128)
plrnn_scan_kernel(const float* __restrict__ AW1, const float* __restrict__ AW2,
                  const float* __restrict__ h1v, const float* __restrict__ h2v,
                  const float* __restrict__ Qv,
                  const float* __restrict__ z0v, const float* __restrict__ a0v,
                  const float* __restrict__ eps, float* __restrict__ Zs)
{
    extern __shared__ float smem[];
    float* W1T  = smem;                // [DZ*129]: W1T[k*129+d] = W1[d][k]
    float* W2T  = W1T + DZ * 129;      // [DZ*129]
    float* vbuf = W2T + DZ * 129;      // [DZ] broadcast vector

    const int d = threadIdx.x;

    // Stage weights: global reads coalesced (row i, col d), LDS writes at
    // d*129+i -> bank (d+i)%64, conflict-free. Diagonal stripped per reference.
    for (int i = 0; i < DZ; ++i) {
        const int idx = i * DZ + d;          // W[i][d]
        const float w1 = AW1[idx];
        const float w2 = AW2[idx];
        W1T[d * 129 + i] = (i == d) ? 0.0f : w1;
        W2T[d * 129 + i] = (i == d) ? 0.0f : w2;
    }

    const float a1d = AW1[d * DZ + d];
    const float a2d = AW2[d * DZ + d];
    const float h1d = h1v[d];
    const float h2d = h2v[d];
    const float Qd  = Qv[d];

    float z = z0v[d];
    float a = a0v[d];

    __syncthreads();

    float eps_cur = eps[d];                               // t = 0
    for (int t = 0; t < TLEN; ++t) {
        // software-pipeline next step's noise load
        float eps_nxt = (t + 1 < TLEN) ? eps[(size_t)(t + 1) * DZ + d] : 0.0f;

        // r = relu(z)
        vbuf[d] = fmaxf(z, 0.0f);
        __syncthreads();

        // mv1 = W1 @ r  (4 independent partials for ILP)
        float s0 = 0.f, s1 = 0.f, s2 = 0.f, s3 = 0.f;
        #pragma unroll 8
        for (int k = 0; k < DZ; k += 4) {
            s0 = fmaf(W1T[(k + 0) * 129 + d], vbuf[k + 0], s0);
            s1 = fmaf(W1T[(k + 1) * 129 + d], vbuf[k + 1], s1);
            s2 = fmaf(W1T[(k + 2) * 129 + d], vbuf[k + 2], s2);
            s3 = fmaf(W1T[(k + 3) * 129 + d], vbuf[k + 3], s3);
        }
        a = fmaf(a1d, a, (s0 + s1) + (s2 + s3));

        const float sg = 1.0f / (1.0f + expf(-z));
        const float bt = sg * (h1d - z);

        __syncthreads();               // mv1 reads done before vbuf reuse
        vbuf[d] = a * bt;
        __syncthreads();

        // mv2 = W2 @ (a*b)
        s0 = 0.f; s1 = 0.f; s2 = 0.f; s3 = 0.f;
        #pragma unroll 8
        for (int k = 0; k < DZ; k += 4) {
            s0 = fmaf(W2T[(k + 0) * 129 + d], vbuf[k + 0], s0);
            s1 = fmaf(W2T[(k + 1) * 129 + d], vbuf[k + 1], s1);
            s2 = fmaf(W2T[(k + 2) * 129 + d], vbuf[k + 2], s2);
            s3 = fmaf(W2T[(k + 3) * 129 + d], vbuf[k + 3], s3);
        }
        z = a2d * z + ((s0 + s1) + (s2 + s3)) + h2d + Qd * eps_cur;

        Zs[(size_t)t * DZ + d] = z;    // coalesced [T, DZ] stream-out
        eps_cur = eps_nxt;
        __syncthreads();               // mv2 reads done before next vbuf write
    }
}

// ---------------------------------------------------------------------------
// Phase 2: X = B @ Z + R .* eta^T, plus transpose-out of Z, using f32 WMMA
// (v_wmma_f32_16x16x4_f32). Block = 128 threads (4 waves) covers 64 timestep
// columns x 128 X-rows (grid.y picks which 128-row half). Each wave owns a
// 16-column stripe; k-outer loop shares one B-fragment across 8 row tiles.
// ---------------------------------------------------------------------------
__global__ void __launch_bounds__(128)
emit_wmma_kernel(const float* __restrict__ Bmat, const float* __restrict__ Rvec,
                 const float* __restrict__ eta,  const float* __restrict__ Zs,
                 float* __restrict__ Xout, float* __restrict__ Zout)
{
    extern __shared__ float smem[];
    float* Bs = smem;                 // [128*130]  Bs[row*130 + k]
    float* Zt = Bs + 128 * 130;       // [128*65]   Zt[k*65 + tcol]

    const int tid   = threadIdx.x;
    const int lane  = tid & 31;
    const int wave  = tid >> 5;           // 0..3
    const int lhalf = lane >> 4;          // 0 | 1
    const int l16   = lane & 15;          // 0..15
    const int tb    = blockIdx.x * 64;    // timestep tile base
    const int rbase = blockIdx.y * 128;   // X-row half base

    // Stage 128 rows of B: coalesced global, conflict-free LDS (pad 130).
    for (int i = 0; i < 128; ++i)
        Bs[i * 130 + tid] = Bmat[(size_t)(rbase + i) * DZ + tid];

    // Stage Z tile: ws is t-major [T,DZ]; LDS is k-major (pad 65 =>
    // write bank (tid+tc)%64, conflict-free). Clamp tail columns for loads.
    for (int tc = 0; tc < 64; ++tc) {
        int t = tb + tc;
        if (t >= TLEN) t = TLEN - 1;
        Zt[tid * 65 + tc] = Zs[(size_t)t * DZ + tid];
    }
    __syncthreads();

    const int tcol = wave * 16;       // tile-local column base of this wave

    v8f acc[8];
    #pragma unroll
    for (int m = 0; m < 8; ++m) acc[m] = {};

    // K loop: 32 steps of K=4. ISA 16x4 f32 A layout: lanes 0-15 carry
    // K=kk+0/kk+1 in the two VGPRs, lanes 16-31 carry K=kk+2/kk+3.
    for (int kk = 0; kk < DZ; kk += 4) {
        const int krow = kk + 2 * lhalf;

        v2f bfrag;                                   // B-matrix = Z tile
        bfrag.x = Zt[(krow + 0) * 65 + tcol + l16];
        bfrag.y = Zt[(krow + 1) * 65 + tcol + l16];

        #pragma unroll
        for (int m = 0; m < 8; ++m) {
            const float* ap = &Bs[(m * 16 + l16) * 130 + krow];
            v2f afrag;                               // A-matrix = B rows
            afrag.x = ap[0];
            afrag.y = ap[1];
            acc[m] = __builtin_amdgcn_wmma_f32_16x16x4_f32(
                false, afrag, false, bfrag, (short)0, acc[m], false, false);
        }
    }

    // Epilogue (divergence only AFTER all WMMAs): fuse R.*eta and store X.
    // C/D layout: VGPR v, lanes 0-15 -> M=v, lanes 16-31 -> M=v+8; N=l16.
    const int t = tb + tcol + l16;
    if (t < TLEN) {
        #pragma unroll
        for (int m = 0; m < 8; ++m) {
            #pragma unroll
            for (int v = 0; v < 8; ++v) {
                const int r = rbase + m * 16 + v + 8 * lhalf;
                const float x =
                    acc[m][v] + Rvec[r] * eta[(size_t)t * DXR + r];
                Xout[(size_t)r * TLEN + t] = x;      // 16 consecutive t / half-wave
            }
        }
    }

    // Transpose-out Z (only row-half 0 blocks). 16 lanes per row => 64B
    // coalesced segments along t; LDS reads conflict-free via pad 65.
    if (blockIdx.y == 0) {
        const int rg = tid >> 4;      // 0..7
        const int tl = tid & 15;      // 0..15
        for (int rr = 0; rr < 16; ++rr) {
            const int dz = rg + rr * 8;              // 0..127
            #pragma unroll
            for (int tc4 = 0; tc4 < 4; ++tc4) {
                const int tc = tl + tc4 * 16;
                const int tt = tb + tc;
                if (tt < TLEN)
                    Zout[(size_t)dz * TLEN + tt] = Zt[dz * 65 + tc];
            }
        }
    }
}

// ---------------------------------------------------------------------------
extern "C" void kernel_launch(void* const* d_in, const int* in_sizes, int n_in,
                              void* d_out, int out_size, void* d_ws, size_t ws_size,
                              hipStream_t stream)
{
    const float* AW1 = (const float*)d_in[0];
    const float* AW2 = (const float*)d_in[1];
    const float* h1  = (const float*)d_in[2];
    const float* h2  = (const float*)d_in[3];
    const float* Bm  = (const float*)d_in[4];
    const float* Q   = (const float*)d_in[5];
    const float* R   = (const float*)d_in[6];
    const float* z0  = (const float*)d_in[7];
    const float* a0  = (const float*)d_in[8];
    const float* eps = (const float*)d_in[9];
    const float* eta = (const float*)d_in[10];

    float* Xout = (float*)d_out;                    // [DX, T]
    float* Zout = Xout + (size_t)DXR * TLEN;        // [DZ, T]
    float* Zs   = (float*)d_ws;                     // scratch [T, DZ]

    const size_t lds1 = (size_t)(2 * DZ * 129 + DZ) * sizeof(float);     // ~129.5 KB
    const size_t lds2 = (size_t)(128 * 130 + 128 * 65) * sizeof(float);  // ~97.5 KB

    // Opt in to >64KB dynamic LDS (WGP has 320KB). Host-side attribute set,
    // deterministic every call, not a stream op (graph-capture safe).
    (void)hipFuncSetAttribute((const void*)plrnn_scan_kernel,
                              hipFuncAttributeMaxDynamicSharedMemorySize, (int)lds1);
    (void)hipFuncSetAttribute((const void*)emit_wmma_kernel,
                              hipFuncAttributeMaxDynamicSharedMemorySize, (int)lds2);

    plrnn_scan_kernel<<<dim3(1), dim3(128), lds1, stream>>>(
        AW1, AW2, h1, h2, Q, z0, a0, eps, Zs);

    dim3 grid((TLEN + 63) / 64, 2);
    emit_wmma_kernel<<<grid, dim3(128), lds2, stream>>>(
        Bm, R, eta, Zs, Xout, Zout);
}